// CropAndResize_1726576857319
// MI455X (gfx1250) — compile-verified
//
#include <hip/hip_runtime.h>

// crop_and_resize for MI455X (gfx1250) — pure HBM-bandwidth-bound gather.
// ~201MB output (NT b128 stores, bypass L2) + <=210MB image reads (L2-resident).
// Per-p sampling tables precomputed in LDS: hot loop = ds_load + 4 gathers +
// lerp + one nontemporal b128 store per 4 outputs. No divisions in hot path.

#define C_DIM    256
#define H_DIM    160
#define W_DIM    160
#define CROP_H   14
#define CROP_W   14
#define CH_SPLIT 8                        // channel slices per box -> 32 ch/block
#define CPB      (C_DIM / CH_SPLIT)       // 32 channels per block
#define THREADS  256
#define PLANE    (H_DIM * W_DIM)          // 25600 floats per channel plane
#define CROP_PIX (CROP_H * CROP_W)        // 196 (divisible by 4)

typedef float v4f __attribute__((ext_vector_type(4)));

__global__ __launch_bounds__(THREADS)
void crop_and_resize_kernel(const float* __restrict__ image,
                            const float* __restrict__ boxes,
                            const int*   __restrict__ box_ind,
                            float*       __restrict__ out,
                            int n_boxes)
{
    // Per-crop-pixel sampling tables (SoA; b32 ds loads, low bank conflict).
    __shared__ int   s_otl[CROP_PIX], s_otr[CROP_PIX];   // tap offsets in plane
    __shared__ int   s_obl[CROP_PIX], s_obr[CROP_PIX];
    __shared__ float s_xl[CROP_PIX],  s_yl[CROP_PIX];
    __shared__ int   s_vv[CROP_PIX];                     // valid flag

    const int bid = blockIdx.x;
    const int box = bid / CH_SPLIT;
    const int c0  = (bid % CH_SPLIT) * CPB;
    if (box >= n_boxes) return;                          // uniform per block

    const int tid = threadIdx.x;

    if (tid < CROP_PIX) {
        // boxes[box] is block-uniform -> scalar loads
        const float y1 = boxes[box * 4 + 0];
        const float x1 = boxes[box * 4 + 1];
        const float y2 = boxes[box * 4 + 2];
        const float x2 = boxes[box * 4 + 3];

        const int p  = tid;
        const int cy = p / CROP_W;          // setup only; not in hot loop
        const int cx = p - cy * CROP_W;

        // y axis (reference arithmetic order)
        float h_scale = (y2 - y1) * (float)(H_DIM - 1) / (float)(CROP_H - 1);
        float in_y    = y1 * (float)(H_DIM - 1) + (float)cy * h_scale;
        int   vy      = (in_y >= 0.0f) && (in_y <= (float)(H_DIM - 1));
        float top     = floorf(in_y);
        float yl      = in_y - top;
        int   ti      = (int)fminf(fmaxf(top,        0.0f), (float)(H_DIM - 1));
        int   bi      = (int)fminf(fmaxf(top + 1.0f, 0.0f), (float)(H_DIM - 1));

        // x axis
        float w_scale = (x2 - x1) * (float)(W_DIM - 1) / (float)(CROP_W - 1);
        float in_x    = x1 * (float)(W_DIM - 1) + (float)cx * w_scale;
        int   vx      = (in_x >= 0.0f) && (in_x <= (float)(W_DIM - 1));
        float left    = floorf(in_x);
        float xl      = in_x - left;
        int   li      = (int)fminf(fmaxf(left,        0.0f), (float)(W_DIM - 1));
        int   ri      = (int)fminf(fmaxf(left + 1.0f, 0.0f), (float)(W_DIM - 1));

        s_otl[p] = ti * W_DIM + li;
        s_otr[p] = ti * W_DIM + ri;
        s_obl[p] = bi * W_DIM + li;
        s_obr[p] = bi * W_DIM + ri;
        s_xl[p]  = xl;
        s_yl[p]  = yl;
        s_vv[p]  = vy && vx;
    }
    __syncthreads();

    const int    b     = box_ind[box];                   // uniform scalar load
    const float* img_b = image + (size_t)b * C_DIM * PLANE;
    float*       out_base = out + ((size_t)box * C_DIM + (size_t)c0) * CROP_PIX;

    const int total4 = (CPB * CROP_PIX) / 4;             // 1568 vec4 outputs

    for (int q = tid; q < total4; q += THREADS) {
        const int idx = q << 2;                          // element index
        const int c   = idx / CROP_PIX;                  // one magic-mul per 4
        const int p0  = idx - c * CROP_PIX;              // 0..192, 4-aligned

        const float* plane = img_b + (size_t)(c0 + c) * PLANE;

        // Speculative prefetch of next channel plane's first tap line
        // (global_prefetch_b8; translation failures silently dropped).
        __builtin_prefetch(plane + PLANE + s_otl[p0], 0, 0);

        float r[4];
#pragma unroll
        for (int j = 0; j < 4; ++j) {
            const int   p  = p0 + j;
            const int   ot = s_otl[p];
            const int   otr= s_otr[p];
            const int   ob = s_obl[p];
            const int   obr= s_obr[p];
            const float xl = s_xl[p];
            const float yl = s_yl[p];
            const bool  vv = (s_vv[p] != 0);

            const float tl = plane[ot];
            const float tr = plane[otr];
            const float bl = plane[ob];
            const float br = plane[obr];

            const float top_v = tl + (tr - tl) * xl;
            const float bot_v = bl + (br - bl) * xl;
            float       val   = top_v + (bot_v - top_v) * yl;
            r[j] = vv ? val : 0.0f;
        }

        // One nontemporal b128 store per 4 outputs (16B-aligned: base and idx
        // are multiples of 16B). Output is write-once; keep L2 for the image.
        v4f v = { r[0], r[1], r[2], r[3] };
        __builtin_nontemporal_store(v, reinterpret_cast<v4f*>(out_base + idx));
    }
}

extern "C" void kernel_launch(void* const* d_in, const int* in_sizes, int n_in,
                              void* d_out, int out_size, void* d_ws, size_t ws_size,
                              hipStream_t stream)
{
    const float* image   = (const float*)d_in[0];
    const float* boxes   = (const float*)d_in[1];
    const int*   box_ind = (const int*)d_in[2];
    float*       out     = (float*)d_out;

    const int n_boxes = in_sizes[1] / 4;                 // boxes is (N,4)
    const int blocks  = n_boxes * CH_SPLIT;

    hipLaunchKernelGGL(crop_and_resize_kernel,
                       dim3(blocks), dim3(THREADS), 0, stream,
                       image, boxes, box_ind, out, n_boxes);
}